// SequenceToSequenceRNN_14688788152960
// MI455X (gfx1250) — compile-verified
//
#include <hip/hip_runtime.h>
#include <hip/hip_bf16.h>

#define DIM   12
#define SEQ   1600
#define PDIM  8
#define BATCH 1024
#define EPSV  1e-5f
// per-layer param block in workspace: Wa[16][12] + Ceff[16][16] + lar[8] + lai[8] + d[16]
#define LAYER_STRIDE (16*12 + 16*16 + 8 + 8 + 16)   // 480 floats

// time chunking for TDM double buffering
#define TCH        64                 // timesteps per chunk
#define NCH        (SEQ / TCH)        // 25 chunks
#define ROWSTRIDE  66                 // padded LDS row stride in floats (64 + 2 pad)
#define TILE_FLOATS (16 * ROWSTRIDE)  // 1056 floats per tile buffer

typedef float v2f __attribute__((ext_vector_type(2)));
typedef float v8f __attribute__((ext_vector_type(8)));
typedef unsigned int v4u __attribute__((ext_vector_type(4)));
typedef int v8i __attribute__((ext_vector_type(8)));
typedef int v4i __attribute__((ext_vector_type(4)));

#if defined(__has_builtin)
#if __has_builtin(__builtin_amdgcn_tensor_load_to_lds) && \
    __has_builtin(__builtin_amdgcn_tensor_store_from_lds) && \
    __has_builtin(__builtin_amdgcn_s_wait_tensorcnt)
#define HAVE_TDM 1
#endif
#endif
#ifndef HAVE_TDM
#define HAVE_TDM 0
#endif

// Swap values between lane L and lane L^16 (SWAPX16 via ds_swizzle group-of-32).
__device__ __forceinline__ float swapx16(float v) {
    return __int_as_float(__builtin_amdgcn_ds_swizzle(__float_as_int(v), 0x401F));
}

__device__ __forceinline__ v2f mk2(float a, float b) {
    v2f t; t.x = a; t.y = b; return t;
}

// D = A(16x4) * B(4x16) + C, fp32 WMMA (CDNA5)
__device__ __forceinline__ v8f wmma4(v2f a, v2f b, v8f c) {
    return __builtin_amdgcn_wmma_f32_16x16x4_f32(
        false, a, false, b, (short)0, c, false, false);
}

#if HAVE_TDM
// Generic LDS pointer -> 32-bit LDS byte offset (low 32 bits of the shared
// aperture address are the LDS offset per the flat-addressing rules).
__device__ __forceinline__ unsigned lds_off(const void* p) {
    return (unsigned)(unsigned long long)(uintptr_t)p;
}

// TDM 2D tile load: tile [16 rows x TCH floats], row stride SEQ floats in
// memory, LDS destination padded to ROWSTRIDE via pad_interval=64dw/pad=2dw.
__device__ __forceinline__ void tdm_load_tile(const float* gsrc, unsigned lds_byte) {
    unsigned long long ga = (unsigned long long)(uintptr_t)gsrc;
    v4u g0;
    g0.x = 1u;                                         // count=1, user descriptor
    g0.y = lds_byte;                                   // lds_addr (bytes)
    g0.z = (unsigned)(ga & 0xFFFFFFFFu);               // global_addr[31:0]
    g0.w = (unsigned)((ga >> 32) & 0x01FFFFFFu) | (2u << 30);  // addr[56:32] | type=2
    v8i g1;
    g1[0] = (2 << 16)      // data_size = 4B
          | (1 << 20)      // pad_enable
          | (5 << 22)      // pad_interval: every 2^(5+1)=64 dwords
          | (1 << 25);     // pad_amount: 2 dwords -> LDS row stride 66
    g1[1] = (int)((unsigned)TCH << 16);                // tensor_dim0[15:0] = TCH
    g1[2] = (int)(16u << 16);                          // tensor_dim1[15:0] = 16 rows
    g1[3] = (int)((unsigned)TCH << 16);                // tile_dim0 = TCH
    g1[4] = 16;                                        // tile_dim1 = 16
    g1[5] = SEQ;                                       // tensor_dim0_stride lo = SEQ
    g1[6] = 0;
    g1[7] = 0;
    v4i g2 = {0, 0, 0, 0};
    v4i g3 = {0, 0, 0, 0};
#if defined(__clang_major__) && (__clang_major__ >= 23)
    v8i gx = {0, 0, 0, 0, 0, 0, 0, 0};
    __builtin_amdgcn_tensor_load_to_lds(g0, g1, g2, g3, gx, 0);
#else
    __builtin_amdgcn_tensor_load_to_lds(g0, g1, g2, g3, 0);
#endif
}

// TDM store with iterate: 16 iterations of a [TCH x 1] row,
// LDS advances by ROWSTRIDE floats, global advances by SEQ floats.
__device__ __forceinline__ void tdm_store_tile(float* gdst, unsigned lds_byte) {
    unsigned long long ga = (unsigned long long)(uintptr_t)gdst;
    v4u g0;
    g0.x = 1u;
    g0.y = lds_byte;
    g0.z = (unsigned)(ga & 0xFFFFFFFFu);
    g0.w = (unsigned)((ga >> 32) & 0x01FFFFFFu) | (2u << 30);
    v8i g1;
    g1[0] = (2 << 16)      // data_size = 4B
          | (1 << 19);     // iterate_enable
    g1[1] = (int)((unsigned)TCH << 16);                // tensor_dim0 = TCH
    g1[2] = (int)(1u << 16);                           // tensor_dim1 = 1
    g1[3] = (int)((unsigned)TCH << 16);                // tile_dim0 = TCH
    g1[4] = 1;                                         // tile_dim1 = 1
    g1[5] = SEQ;
    g1[6] = 0;
    g1[7] = 0;
    v4i g2;
    g2[0] = 0;                                         // tensor_dim2
    g2[1] = ROWSTRIDE;                                 // lds_addr_increment (elems)
    g2[2] = SEQ;                                       // global_addr_increment lo
    g2[3] = (16 - 1) << 16;                            // iterate_count = 15 (16x)
    v4i g3 = {0, 0, 0, 0};
#if defined(__clang_major__) && (__clang_major__ >= 23)
    v8i gx = {0, 0, 0, 0, 0, 0, 0, 0};
    __builtin_amdgcn_tensor_store_from_lds(g0, g1, g2, g3, gx, 0);
#else
    __builtin_amdgcn_tensor_store_from_lds(g0, g1, g2, g3, 0);
#endif
}
#endif  // HAVE_TDM

// ---------------------------------------------------------------------------
// Kernel 1: per-seq-position BatchNorm scale/shift.
// out[b,l,h] = x[b,l]*w[h]+b[h]  ->  stats over (b,h) reduce to batch stats of
// x combined with scalar stats of (w,b).
// ---------------------------------------------------------------------------
__global__ __launch_bounds__(256)
void bn_stats_kernel(const float* __restrict__ x,
                     const float* __restrict__ l1w, const float* __restrict__ l1b,
                     const float* __restrict__ gamma, const float* __restrict__ beta,
                     float* __restrict__ scale, float* __restrict__ shift) {
    const int l = blockIdx.x;
    const int t = threadIdx.x;
    __shared__ float s1[256];
    __shared__ float s2[256];
    float sx = 0.f, sx2 = 0.f;
    for (int b = t; b < BATCH; b += 256) {
        float v = x[(size_t)b * SEQ + l];
        sx += v; sx2 += v * v;
    }
    s1[t] = sx; s2[t] = sx2;
    __syncthreads();
    for (int off = 128; off > 0; off >>= 1) {
        if (t < off) { s1[t] += s1[t + off]; s2[t] += s2[t + off]; }
        __syncthreads();
    }
    if (t == 0) {
        float mw = 0.f, mb = 0.f, sw2 = 0.f, swb = 0.f, sb2 = 0.f;
        for (int h = 0; h < DIM; ++h) {
            float w = l1w[h], bb = l1b[h];
            mw += w; mb += bb; sw2 += w * w; swb += w * bb; sb2 += bb * bb;
        }
        const float inv = 1.0f / (float)DIM;
        mw *= inv; mb *= inv; sw2 *= inv; swb *= inv; sb2 *= inv;
        float mx  = s1[0] / (float)BATCH;
        float mx2 = s2[0] / (float)BATCH;
        float mean = mx * mw + mb;
        float e2   = mx2 * sw2 + 2.f * mx * swb + sb2;
        float var  = e2 - mean * mean;
        float sc   = gamma[l] * rsqrtf(var + EPSV);
        scale[l] = sc;
        shift[l] = beta[l] - mean * sc;
    }
}

// ---------------------------------------------------------------------------
// Kernel 2: ZOH discretization + WMMA operand tables. Row index m = 2p + ri.
// ---------------------------------------------------------------------------
__global__ __launch_bounds__(64)
void s5_prep_kernel(const float* __restrict__ lam_re, const float* __restrict__ lam_im,
                    const float* __restrict__ b_re,   const float* __restrict__ b_im,
                    const float* __restrict__ c_re,   const float* __restrict__ c_im,
                    const float* __restrict__ dvec,   const float* __restrict__ log_step,
                    float* __restrict__ lbase) {
    const int t = threadIdx.x;
    const int layer = t >> 4;
    const int m = t & 15;
    if (layer >= 3) return;
    const int p  = m >> 1;
    const int ri = m & 1;

    float stp = __expf(log_step[layer * PDIM + p]);
    float lr  = lam_re[layer * PDIM + p];
    float li  = lam_im[layer * PDIM + p];
    float er  = __expf(lr * stp);
    float lbr = er * __cosf(li * stp);
    float lbi = er * __sinf(li * stp);
    float den = lr * lr + li * li;
    float nr = lbr - 1.0f, ni = lbi;
    float cr = (nr * lr + ni * li) / den;
    float ci = (ni * lr - nr * li) / den;

    float* base = lbase + layer * LAYER_STRIDE;
    float* Wa   = base;                 // 16*12
    float* Ceff = base + 192;           // 16*16
    float* plar = base + 192 + 256;     // 8
    float* plai = plar + 8;             // 8
    float* pdd  = plai + 8;             // 16

    for (int h = 0; h < DIM; ++h) {
        float br = b_re[(layer * PDIM + p) * DIM + h];
        float bi = b_im[(layer * PDIM + p) * DIM + h];
        float vr = cr * br - ci * bi;
        float vi = cr * bi + ci * br;
        Wa[m * DIM + h] = ri ? vi : vr;
    }
    if (ri == 0) { plar[p] = lbr; plai[p] = lbi; }

    for (int k = 0; k < 16; ++k) {
        int pp = k >> 1;
        float v = 0.f;
        if (m < DIM)
            v = (k & 1) ? -2.f * c_im[(layer * DIM + m) * PDIM + pp]
                        :  2.f * c_re[(layer * DIM + m) * PDIM + pp];
        Ceff[m * 16 + k] = v;
    }
    pdd[m] = (m < DIM) ? dvec[layer * DIM + m] : 0.f;
}

// ---------------------------------------------------------------------------
// Kernel 3: fused l1 + BN + 3x S5 + l2, one wave32 per 16 batch columns.
// x is staged to LDS by TDM (double buffered), outputs drain via TDM store.
// ---------------------------------------------------------------------------
__global__ __launch_bounds__(64)
void s5_fused_kernel(const float* __restrict__ x, const float* __restrict__ ws,
                     const float* __restrict__ l1w, const float* __restrict__ l1b,
                     const float* __restrict__ l2w, const float* __restrict__ l2b,
                     float* __restrict__ out) {
#if HAVE_TDM
    __shared__ float smem[2 /*wave*/][2 /*x,out*/][2 /*dbuf*/][TILE_FLOATS];
#endif
    const int lane = threadIdx.x & 31;
    const int wave = threadIdx.x >> 5;
    const int tile = blockIdx.x * 2 + wave;
    const int half = (lane >> 4) & 1;
    const int m    = lane & 15;           // A row; B/C/D column (batch)
    const int b0   = tile * 16;           // first batch of this wave's tile
    const int b    = b0 + m;

    const float* bnscale = ws;
    const float* bnshift = ws + SEQ;
    const float* lbase   = ws + 2 * SEQ;

    float wl1[8], bl1[8], w2l[8], umask[8];
#pragma unroll
    for (int r = 0; r < 8; ++r) {
        int h = 8 * half + r;
        bool valid = (h < DIM);
        wl1[r]   = valid ? l1w[h] : 0.f;
        bl1[r]   = valid ? l1b[h] : 0.f;
        w2l[r]   = valid ? l2w[h] : 0.f;
        umask[r] = valid ? 1.f : 0.f;
    }
    const float bias2 = l2b[0];

    v2f waA[3][3];
    v2f ceA[3][4];
    float lar[3][4], lai[3][4], dd[3][8];
#pragma unroll
    for (int L = 0; L < 3; ++L) {
        const float* base = lbase + L * LAYER_STRIDE;
        const float* Wa = base;
        const float* Ce = base + 192;
        const float* pr = base + 192 + 256;
        const float* pi = pr + 8;
        const float* pd = pi + 8;
#pragma unroll
        for (int j = 0; j < 3; ++j) {
            int k0 = 4 * j + 2 * half;
            waA[L][j] = mk2(Wa[m * DIM + k0], Wa[m * DIM + k0 + 1]);
        }
#pragma unroll
        for (int j = 0; j < 4; ++j) {
            int k0 = 4 * j + 2 * half;
            ceA[L][j] = mk2(Ce[m * 16 + k0], Ce[m * 16 + k0 + 1]);
        }
#pragma unroll
        for (int k = 0; k < 4; ++k) {
            int p = k + 4 * half;
            lar[L][k] = pr[p];
            lai[L][k] = pi[p];
        }
#pragma unroll
        for (int r = 0; r < 8; ++r) dd[L][r] = pd[8 * half + r];
    }

    float st[3][8];
#pragma unroll
    for (int L = 0; L < 3; ++L)
#pragma unroll
        for (int r = 0; r < 8; ++r) st[L][r] = 0.f;

#if HAVE_TDM
    unsigned xoff[2] = { lds_off(&smem[wave][0][0][0]), lds_off(&smem[wave][0][1][0]) };
    unsigned ooff[2] = { lds_off(&smem[wave][1][0][0]), lds_off(&smem[wave][1][1][0]) };
    // prime the pipeline: chunk 0 load
    tdm_load_tile(x + (size_t)b0 * SEQ, xoff[0]);
#endif

    for (int c = 0; c < NCH; ++c) {
        const int cur = c & 1;
#if HAVE_TDM
        if (c + 1 < NCH)
            tdm_load_tile(x + (size_t)b0 * SEQ + (size_t)(c + 1) * TCH, xoff[(c + 1) & 1]);
        // TDM ops complete in order per wave.  Newer-than-L(c) ops still
        // outstanding: S(c-1) (if c>0) and L(c+1) (if it exists).
        if (c == 0 || c == NCH - 1) __builtin_amdgcn_s_wait_tensorcnt(1);
        else                        __builtin_amdgcn_s_wait_tensorcnt(2);
        __asm__ volatile("" ::: "memory");
        volatile const float* xtile = &smem[wave][0][cur][0];
        volatile float*       otile = &smem[wave][1][cur][0];
#endif
        if (c + 1 < NCH) __builtin_prefetch(bnscale + (c + 1) * TCH, 0, 0);

        for (int tt = 0; tt < TCH; ++tt) {
            const int l = c * TCH + tt;
#if HAVE_TDM
            float xl = xtile[m * ROWSTRIDE + tt];     // broadcast across halves
#else
            float xl = x[(size_t)b * SEQ + l];
#endif
            float sc = bnscale[l];
            float sh = bnshift[l];

            float res[8], u[8], yout[8];
#pragma unroll
            for (int r = 0; r < 8; ++r) {
                res[r] = xl * wl1[r] + bl1[r];               // pre-BN residual
                u[r]   = umask[r] * (res[r] * sc + sh);      // BN'd layer input
            }

#pragma unroll
            for (int L = 0; L < 3; ++L) {
                // input projection: Bu(16x16) = Wa(16x12) * u(12x16)
                float tu[8];
#pragma unroll
                for (int r = 0; r < 8; ++r) tu[r] = swapx16(u[r]);
                v2f bu0 = half ? mk2(tu[2], tu[3]) : mk2(u[0], u[1]);
                v2f bu1 = half ? mk2(tu[6], tu[7]) : mk2(u[4], u[5]);
                v2f bu2 = half ? mk2(u[2], u[3])   : mk2(tu[0], tu[1]);
                v8f bu = {};
                bu = wmma4(waA[L][0], bu0, bu);
                bu = wmma4(waA[L][1], bu1, bu);
                bu = wmma4(waA[L][2], bu2, bu);

                // recurrence: s' = lam_bar * s + Bu (complex, in-lane)
#pragma unroll
                for (int k = 0; k < 4; ++k) {
                    float sr = st[L][2 * k], si = st[L][2 * k + 1];
                    float nr = lar[L][k] * sr - lai[L][k] * si + bu[2 * k];
                    float ni = lar[L][k] * si + lai[L][k] * sr + bu[2 * k + 1];
                    st[L][2 * k] = nr; st[L][2 * k + 1] = ni;
                }

                // output projection: y(16x16) = Ceff(16x16) * S(16x16)
                float ts[8];
#pragma unroll
                for (int r = 0; r < 8; ++r) ts[r] = swapx16(st[L][r]);
                v2f bs0 = half ? mk2(ts[2], ts[3])       : mk2(st[L][0], st[L][1]);
                v2f bs1 = half ? mk2(ts[6], ts[7])       : mk2(st[L][4], st[L][5]);
                v2f bs2 = half ? mk2(st[L][2], st[L][3]) : mk2(ts[0], ts[1]);
                v2f bs3 = half ? mk2(st[L][6], st[L][7]) : mk2(ts[4], ts[5]);
                v8f y = {};
                y = wmma4(ceA[L][0], bs0, y);
                y = wmma4(ceA[L][1], bs1, y);
                y = wmma4(ceA[L][2], bs2, y);
                y = wmma4(ceA[L][3], bs3, y);

#pragma unroll
                for (int r = 0; r < 8; ++r) yout[r] = y[r] + dd[L][r] * u[r];

                if (L < 2) {
#pragma unroll
                    for (int r = 0; r < 8; ++r) {
                        float v = fmaxf(yout[r], 0.f) + res[r];   // relu + residual
                        u[r] = v; res[r] = v;
                    }
                }
            }

            // final Linear(12 -> 1): dot over h, folded across lane halves
            float acc = 0.f;
#pragma unroll
            for (int r = 0; r < 8; ++r) acc += yout[r] * w2l[r];
            acc += swapx16(acc);
#if HAVE_TDM
            if (half == 0) otile[m * ROWSTRIDE + tt] = acc + bias2;
#else
            if (half == 0) out[(size_t)b * SEQ + l] = acc + bias2;
#endif
        }

#if HAVE_TDM
        // make this chunk's LDS writes visible before the TDM store reads them
        __asm__ volatile("s_wait_dscnt 0" ::: "memory");
        tdm_store_tile(out + (size_t)b0 * SEQ + (size_t)c * TCH, ooff[cur]);
#endif
    }

#if HAVE_TDM
    __builtin_amdgcn_s_wait_tensorcnt(0);
#endif
}

extern "C" void kernel_launch(void* const* d_in, const int* in_sizes, int n_in,
                              void* d_out, int out_size, void* d_ws, size_t ws_size,
                              hipStream_t stream) {
    const float* x     = (const float*)d_in[0];
    const float* l1w   = (const float*)d_in[1];
    const float* l1b   = (const float*)d_in[2];
    const float* l2w   = (const float*)d_in[3];
    const float* l2b   = (const float*)d_in[4];
    const float* gamma = (const float*)d_in[5];
    const float* beta  = (const float*)d_in[6];
    const float* lre   = (const float*)d_in[7];
    const float* lim   = (const float*)d_in[8];
    const float* bre   = (const float*)d_in[9];
    const float* bim   = (const float*)d_in[10];
    const float* cre   = (const float*)d_in[11];
    const float* cim   = (const float*)d_in[12];
    const float* dvec  = (const float*)d_in[13];
    const float* lstep = (const float*)d_in[14];
    float* out = (float*)d_out;

    float* ws    = (float*)d_ws;
    float* scale = ws;                 // [SEQ]
    float* shift = ws + SEQ;           // [SEQ]
    float* lbase = ws + 2 * SEQ;       // 3 * LAYER_STRIDE

    bn_stats_kernel<<<SEQ, 256, 0, stream>>>(x, l1w, l1b, gamma, beta, scale, shift);
    s5_prep_kernel<<<1, 64, 0, stream>>>(lre, lim, bre, bim, cre, cim, dvec, lstep, lbase);
    // 1024 batch / 16 per wave / 2 waves per block = 32 blocks x 64 threads
    s5_fused_kernel<<<BATCH / 32, 64, 0, stream>>>(x, ws, l1w, l1b, l2w, l2b, out);
}